// IHCPaulick2024_4148938408289
// MI455X (gfx1250) — compile-verified
//
#include <hip/hip_runtime.h>
#include <stdint.h>

// ---------------------------------------------------------------------------
// IHC Paulick2024: elementwise MET conductance + affine linear recurrence,
// parallelized as a 3-pass associative scan over (a,b) affine maps.
// CDNA5 paths: global_load_async_to_lds_b128 + s_wait_asynccnt (ASYNCcnt
// double-buffered staging), wave32 shuffles, swizzled ds_load_b128,
// NT output stores, L2-aware reversed block order in the solve pass.
// ---------------------------------------------------------------------------

typedef __attribute__((ext_vector_type(4))) float v4f;   // native vector for b128

namespace ihc {
constexpr double dTsC  = (1.0 / 44100.0) / 12.5e-12;           // Ts / CM
constexpr float  c_Ts    = (float)dTsC;
constexpr float  c_SCALE = 5.623413251903491e-06f;              // 10^(-105/20)
constexpr float  c_X0    = 20e-9f;
constexpr float  c_invS0 = (float)(1.0 / 16e-9);
constexpr float  c_invS1 = (float)(1.0 / 35e-9);
constexpr float  c_GMET  = 30e-9f;
constexpr float  c_GKF   = 19.8e-9f;
constexpr float  c_GKS   = 19.8e-9f;
constexpr float  c_EP    = 90e-3f;
constexpr float  c_EKF   = -71e-3f;
constexpr float  c_EKS   = -78e-3f;
constexpr float  c_VREST = -57.03e-3f;
constexpr float  c_GPRE  = 3.3514e-9f;
constexpr int    NPRE    = 2205;                                // 44100*0.05

// v' = a*v + b with a = c_aC - G*c_Ts, b = G*c_bEP + c_bKE
constexpr float  c_aC  = (float)(1.0 - (19.8e-9 + 19.8e-9) * dTsC);
constexpr float  c_bEP = (float)(90e-3 * dTsC);
constexpr float  c_bKE = (float)((19.8e-9 * -71e-3 + 19.8e-9 * -78e-3) * dTsC);

constexpr int T      = 131072;
constexpr int LANES  = 400;              // 8 * 50
constexpr int CH     = 2048;             // samples per chunk
constexpr int NCHUNK = T / CH;           // 64 chunks per lane
constexpr int NCHTOT = LANES * NCHUNK;   // 25600 chunks total
constexpr int TPB    = 256;              // 8 wave32
constexpr int KPT    = CH / TPB;         // 8 samples per thread
constexpr int NWAVE  = TPB / 32;
constexpr int CPB    = 4;                // chunks per block in solve pass
constexpr int ROWF   = CH + 128;         // padded LDS row (floats)
} // namespace ihc

using namespace ihc;

// --- helpers ---------------------------------------------------------------

__device__ __forceinline__ uint32_t lds_off(const void* p) {
    // generic pointer to LDS: low 32 bits are the workgroup-relative LDS addr
    return (uint32_t)(uintptr_t)p;
}

__device__ __forceinline__ void async_b128(uint32_t ldsoff, uint32_t voff,
                                           const float* sbase) {
    // GLOBAL_LOAD_ASYNC_TO_LDS_B128 (GVS mode): copies 16 bytes from global sbase plus offset voff to the LDS address in ldsoff
    asm volatile("global_load_async_to_lds_b128 %0, %1, %2"
                 :: "v"(ldsoff), "v"(voff), "s"(sbase)
                 : "memory");
}

// Per-sample affine map from BM velocity x:  v' = a*v + b
__device__ __forceinline__ void met_ab(float xv, float& a, float& b) {
    float z  = c_X0 - c_SCALE * xv;
    float e0 = __expf(z * c_invS0);
    float e1 = __expf(z * c_invS1);
    float den = fmaf(e1, 1.0f + e0, 1.0f);       // 1 + e1*(1+e0)  (inf-safe)
    float G   = c_GMET / den;
    a = c_aC - G * c_Ts;
    b = fmaf(G, c_bEP, c_bKE);
}

// wave32 inclusive scan of ordered affine maps (lane order = time order)
__device__ __forceinline__ void wave_incl_scan(float& A, float& B, int lane) {
#pragma unroll
    for (int d = 1; d < 32; d <<= 1) {
        float Ae = __shfl_up(A, d, 32);
        float Be = __shfl_up(B, d, 32);
        if (lane >= d) { B = fmaf(A, Be, B); A *= Ae; }
    }
}

// --- pass 0: scalar pre-charge (compiles to SALU float on gfx1250) --------

__global__ void k_precharge(float* __restrict__ vpre) {
    float v = c_VREST;
    for (int i = 0; i < NPRE; ++i) {
        float Im = c_GPRE * (v - c_EP);
        float Ik = c_GKF  * (v - c_EKF);
        float Is = c_GKS  * (v - c_EKS);
        v = v - (Im + Ik + Is) * c_Ts;
    }
    vpre[0] = v;
}

// --- pass 1: per-chunk affine aggregate (A,B), [chunk][lane] layout -------

__global__ __launch_bounds__(TPB) void
k_aggregate(const float* __restrict__ x,
            float* __restrict__ wsA, float* __restrict__ wsB) {
    const int cid = blockIdx.x;                  // chunks tile (lane,T) exactly
    const int t   = threadIdx.x;
    const size_t base = (size_t)cid * CH + (size_t)t * KPT;

    const v4f x0 = ((const v4f*)(x + base))[0];
    const v4f x1 = ((const v4f*)(x + base))[1];
    const float xs[KPT] = {x0.x, x0.y, x0.z, x0.w, x1.x, x1.y, x1.z, x1.w};

    float A = 1.0f, B = 0.0f;
#pragma unroll
    for (int k = 0; k < KPT; ++k) {
        float a, b; met_ab(xs[k], a, b);
        B = fmaf(a, B, b);                       // new element applied after prefix
        A = a * A;
    }

    const int lane = t & 31, w = t >> 5;
    wave_incl_scan(A, B, lane);

    __shared__ float sA[NWAVE], sB[NWAVE];
    if (lane == 31) { sA[w] = A; sB[w] = B; }
    __syncthreads();
    if (t == 0) {
        float Ab = sA[0], Bb = sB[0];
#pragma unroll
        for (int j = 1; j < NWAVE; ++j) { Bb = fmaf(sA[j], Bb, sB[j]); Ab = sA[j] * Ab; }
        const int l = cid / NCHUNK, c = cid % NCHUNK;
        wsA[c * LANES + l] = Ab;                 // transposed: coalesced in pass 2
        wsB[c * LANES + l] = Bb;
    }
}

// --- pass 2: serial scan of chunk aggregates per lane (coalesced) ---------

__global__ void k_chunkscan(const float* __restrict__ wsA,
                            const float* __restrict__ wsB,
                            const float* __restrict__ vpre,
                            float* __restrict__ vstart) {
    const int l = blockIdx.x * blockDim.x + threadIdx.x;
    if (l >= LANES) return;
    float v = vpre[0];
    for (int c = 0; c < NCHUNK; ++c) {
        vstart[c * LANES + l] = v;
        v = fmaf(wsA[c * LANES + l], v, wsB[c * LANES + l]);
    }
}

// --- pass 3: solve + write output, async double-buffered staging ----------
// Block order reversed: pass 1 leaves the TAIL of x resident in the 192MB L2,
// so pass 3 consumes the tail first (L2 hits) before the evicted head.

__global__ __launch_bounds__(TPB) void
k_solve(const float* __restrict__ x,
        const float* __restrict__ vstart,
        const float* __restrict__ vprep,
        float* __restrict__ out) {
    __shared__ float smem[2][ROWF];              // padded for bank swizzle
    __shared__ float sA[NWAVE], sB[NWAVE];

    const int t    = threadIdx.x;
    const int lane = t & 31, w = t >> 5;
    const int bg   = (int)gridDim.x - 1 - (int)blockIdx.x;   // reversed order
    const int cid0 = bg * CPB;

    // swizzled LDS byte offset: +16B per group of 8 threads -> 2-way optimum
    const uint32_t soff = (uint32_t)(t * (KPT * 4) + ((t >> 3) << 4));
    const uint32_t lds0 = lds_off(&smem[0][0]) + soff;
    const uint32_t lds1 = lds_off(&smem[1][0]) + soff;
    const uint32_t voff = (uint32_t)(t * (KPT * 4));

    // prefetch chunk 0 into buffer 0 (each thread stages its own 32 B)
    {
        const float* gx = x + (size_t)cid0 * CH;
        async_b128(lds0,      voff,      gx);
        async_b128(lds0 + 16, voff + 16, gx);
    }
    const float vpre = vprep[0];

    for (int ci = 0; ci < CPB; ++ci) {
        const int cid = cid0 + ci;

        if (ci + 1 < CPB) {                      // prefetch next chunk
            const float* gx = x + (size_t)(cid + 1) * CH;
            const uint32_t ldsn = ((ci + 1) & 1) ? lds1 : lds0;
            async_b128(ldsn,      voff,      gx);
            async_b128(ldsn + 16, voff + 16, gx);
            asm volatile("s_wait_asynccnt 2" ::: "memory");  // current chunk done
        } else {
            asm volatile("s_wait_asynccnt 0" ::: "memory");
        }

        // read back the 8 samples this thread staged (ds_load_b128 x2)
        const int buf = ci & 1;
        const v4f* sp = (const v4f*)((const char*)&smem[buf][0] + soff);
        const v4f s0 = sp[0];
        const v4f s1 = sp[1];
        const float xs[KPT] = {s0.x, s0.y, s0.z, s0.w, s1.x, s1.y, s1.z, s1.w};

        float a[KPT], b[KPT];
        float At = 1.0f, Bt = 0.0f;
#pragma unroll
        for (int k = 0; k < KPT; ++k) {
            met_ab(xs[k], a[k], b[k]);
            Bt = fmaf(a[k], Bt, b[k]);
            At = a[k] * At;
        }

        // block-level exclusive affine prefix for this thread
        float Ai = At, Bi = Bt;
        wave_incl_scan(Ai, Bi, lane);
        if (lane == 31) { sA[w] = Ai; sB[w] = Bi; }
        __syncthreads();
        float Awx = 1.0f, Bwx = 0.0f;            // prefix of earlier waves
        for (int j = 0; j < w; ++j) { Bwx = fmaf(sA[j], Bwx, sB[j]); Awx = sA[j] * Awx; }
        __syncthreads();                         // sA/sB reused next iter

        float Ael = __shfl_up(Ai, 1, 32);        // thread-exclusive in wave
        float Bel = __shfl_up(Bi, 1, 32);
        if (lane == 0) { Ael = 1.0f; Bel = 0.0f; }
        const float Ax = Ael * Awx;              // Ael ∘ Awx
        const float Bx = fmaf(Ael, Bwx, Bel);

        // starting membrane potential for this thread's 8 samples
        const int l_ = cid / NCHUNK, c_ = cid % NCHUNK;
        float v = fmaf(Ax, vstart[c_ * LANES + l_], Bx);

        float o[KPT];
#pragma unroll
        for (int k = 0; k < KPT; ++k) {
            v = fmaf(a[k], v, b[k]);
            o[k] = v - vpre;
        }
        // NT stores: output is write-once; don't evict x from L2
        v4f* op = (v4f*)(out + (size_t)cid * CH + (size_t)t * KPT);
        const v4f o0 = {o[0], o[1], o[2], o[3]};
        const v4f o1 = {o[4], o[5], o[6], o[7]};
        __builtin_nontemporal_store(o0, op);
        __builtin_nontemporal_store(o1, op + 1);
    }
}

// --- launch ---------------------------------------------------------------

extern "C" void kernel_launch(void* const* d_in, const int* in_sizes, int n_in,
                              void* d_out, int out_size, void* d_ws, size_t ws_size,
                              hipStream_t stream) {
    (void)in_sizes; (void)n_in; (void)out_size; (void)ws_size;
    const float* x   = (const float*)d_in[0];
    float*       out = (float*)d_out;

    float* ws   = (float*)d_ws;
    float* vpre = ws;                   // [0]
    float* wsA  = ws + 64;              // [NCHTOT]
    float* wsB  = wsA + NCHTOT;         // [NCHTOT]
    float* vst  = wsB + NCHTOT;         // [NCHTOT]   (~308 KB total)

    k_precharge<<<1, 1, 0, stream>>>(vpre);
    k_aggregate<<<NCHTOT, TPB, 0, stream>>>(x, wsA, wsB);
    k_chunkscan<<<(LANES + TPB - 1) / TPB, TPB, 0, stream>>>(wsA, wsB, vpre, vst);
    k_solve<<<NCHTOT / CPB, TPB, 0, stream>>>(x, vst, vpre, out);
}